// Model_22522808500707
// MI455X (gfx1250) — compile-verified
//
#include <hip/hip_runtime.h>

#define H 128
#define ND_ 100000
#define NA_ 50000
#define E_  500000
#define EL_ 200000

typedef __attribute__((ext_vector_type(2))) float v2f;
typedef __attribute__((ext_vector_type(8))) float v8f;

// ---------------------------------------------------------------------------
// WMMA f32 16x16x4 GEMM:  C[M x 128] = act(A1@W1 (+ A2@W2) + bias)
// One block = 16 output rows, 8 waves cover the 8 N-tiles of width 16.
// A fragment (16x4 f32, 2 VGPRs/lane): lanes 0-15 hold K=k0,k0+1 for M=lane;
// lanes 16-31 hold K=k0+2,k0+3 for M=lane-16.
// B fragment (4x16 f32): v0/v1 hold rows kb,kb+1, N = lane&15, kb = k0+2*(lane>=16).
// C/D: VGPR r -> M = r + 8*(lane>=16), N = lane&15.
// ---------------------------------------------------------------------------
__device__ __forceinline__ void gemm_pass(const float* __restrict__ A,
                                          const float* __restrict__ W,
                                          int m0, int n0, int lane, v8f& acc) {
  const int row  = m0 + (lane & 15);
  const int half = lane >> 4;                 // 0 or 1
  const int col  = n0 + (lane & 15);
  const float* __restrict__ Arow = A + (size_t)row * H;
#pragma unroll
  for (int k0 = 0; k0 < H; k0 += 4) {
    const int kb = k0 + 2 * half;
    v2f a;
    a.x = Arow[kb];
    a.y = Arow[kb + 1];
    v2f b;
    b.x = W[(size_t)kb * H + col];
    b.y = W[(size_t)(kb + 1) * H + col];
    acc = __builtin_amdgcn_wmma_f32_16x16x4_f32(
        /*neg_a=*/false, a, /*neg_b=*/false, b,
        /*c_mod=*/(short)0, acc, /*reuse_a=*/false, /*reuse_b=*/false);
  }
}

__global__ __launch_bounds__(256) void gemm_dual(const float* __restrict__ A1,
                                                 const float* __restrict__ W1,
                                                 const float* __restrict__ A2,
                                                 const float* __restrict__ W2,
                                                 const float* __restrict__ bias,
                                                 float* __restrict__ C, int relu) {
  const int lane = threadIdx.x & 31;
  const int wave = threadIdx.x >> 5;
  const int m0 = blockIdx.x * 16;
  const int n0 = wave * 16;
  const int col = n0 + (lane & 15);

  v8f acc;
  const float bv = bias[col];
#pragma unroll
  for (int i = 0; i < 8; ++i) acc[i] = bv;

  gemm_pass(A1, W1, m0, n0, lane, acc);
  if (A2) gemm_pass(A2, W2, m0, n0, lane, acc);

  const int rbase = m0 + 8 * (lane >> 4);
#pragma unroll
  for (int r = 0; r < 8; ++r) {
    float v = acc[r];
    if (relu) v = fmaxf(v, 0.0f);
    C[(size_t)(rbase + r) * H + col] = v;
  }
}

// ---------------------------------------------------------------------------
// Edge scatter: agg_a[dst] += xd[src];  agg_d[src] += xa[dst];  (+ degrees)
// One block (128 threads) per edge, one channel per thread.
// ---------------------------------------------------------------------------
__device__ __forceinline__ void atomic_add_f32(float* p, float v) {
  __hip_atomic_fetch_add(p, v, __ATOMIC_RELAXED, __HIP_MEMORY_SCOPE_AGENT);
}

__global__ __launch_bounds__(128) void scatter_edges(const float* __restrict__ xd,
                                                     const float* __restrict__ xa,
                                                     const int* __restrict__ src,
                                                     const int* __restrict__ dst,
                                                     float* __restrict__ agg_a,
                                                     float* __restrict__ agg_d,
                                                     float* deg_a, float* deg_d, int nE) {
  const int e = blockIdx.x;
  if (e >= nE) return;
  const int t = threadIdx.x;
  const int s = src[e];
  const int d = dst[e];
  atomic_add_f32(&agg_a[(size_t)d * H + t], xd[(size_t)s * H + t]);
  atomic_add_f32(&agg_d[(size_t)s * H + t], xa[(size_t)d * H + t]);
  if (deg_a && t == 0) atomic_add_f32(&deg_a[d], 1.0f);
  if (deg_d && t == 1) atomic_add_f32(&deg_d[s], 1.0f);
}

__global__ void normalize_agg(float* __restrict__ agg, const float* __restrict__ deg, int M) {
  const size_t i = (size_t)blockIdx.x * blockDim.x + threadIdx.x;
  if (i >= (size_t)M * H) return;
  const float d = deg[i >> 7];
  agg[i] = agg[i] / fmaxf(d, 1.0f);
}

__global__ void gather_author(const float* __restrict__ emb, const int* __restrict__ nid,
                              float* __restrict__ out, int M) {
  const size_t i = (size_t)blockIdx.x * blockDim.x + threadIdx.x;
  if (i >= (size_t)M * H) return;
  const int r = (int)(i >> 7);
  const int c = (int)(i & (H - 1));
  out[i] = emb[(size_t)nid[r] * H + c];
}

// ---------------------------------------------------------------------------
// Edge scorer: pred[e] = dot(o_d[i], o_a[j]) over H=128; one wave per edge.
// ---------------------------------------------------------------------------
__global__ __launch_bounds__(256) void score_edges(const float* __restrict__ od,
                                                   const float* __restrict__ oa,
                                                   const int* __restrict__ eli,
                                                   float* __restrict__ pred, int nEL) {
  const int wave = threadIdx.x >> 5;
  const int lane = threadIdx.x & 31;
  const int e = blockIdx.x * 8 + wave;
  if (e >= nEL) return;
  const int i = eli[e];
  const int j = eli[nEL + e];
  const float4 a = ((const float4*)(od + (size_t)i * H))[lane];
  const float4 b = ((const float4*)(oa + (size_t)j * H))[lane];
  float s = a.x * b.x + a.y * b.y + a.z * b.z + a.w * b.w;
#pragma unroll
  for (int off = 16; off > 0; off >>= 1) s += __shfl_xor(s, off, 32);
  if (lane == 0) pred[e] = s;
}

// ---------------------------------------------------------------------------
extern "C" void kernel_launch(void* const* d_in, const int* in_sizes, int n_in,
                              void* d_out, int out_size, void* d_ws, size_t ws_size,
                              hipStream_t stream) {
  const float* doc_x      = (const float*)d_in[0];
  const int*   author_nid = (const int*)d_in[1];
  const int*   edge_index = (const int*)d_in[2];
  const int*   eli        = (const int*)d_in[3];
  const float* doc_lin_W  = (const float*)d_in[4];
  const float* doc_lin_b  = (const float*)d_in[5];
  const float* author_emb = (const float*)d_in[6];
  const float* c1_da_Wl = (const float*)d_in[7];
  const float* c1_da_bl = (const float*)d_in[8];
  const float* c1_da_Wr = (const float*)d_in[9];
  const float* c1_ad_Wl = (const float*)d_in[10];
  const float* c1_ad_bl = (const float*)d_in[11];
  const float* c1_ad_Wr = (const float*)d_in[12];
  const float* c2_da_Wl = (const float*)d_in[13];
  const float* c2_da_bl = (const float*)d_in[14];
  const float* c2_da_Wr = (const float*)d_in[15];
  const float* c2_ad_Wl = (const float*)d_in[16];
  const float* c2_ad_bl = (const float*)d_in[17];
  const float* c2_ad_Wr = (const float*)d_in[18];

  const int* src = edge_index;        // doc side
  const int* dst = edge_index + E_;   // author side

  // workspace layout (floats)
  float* ws     = (float*)d_ws;
  float* x_doc  = ws;
  float* x_auth = x_doc  + (size_t)ND_ * H;
  float* agg_a  = x_auth + (size_t)NA_ * H;
  float* agg_d  = agg_a  + (size_t)NA_ * H;
  float* h_a    = agg_d  + (size_t)ND_ * H;
  float* h_d    = h_a    + (size_t)NA_ * H;
  float* o_a    = h_d    + (size_t)ND_ * H;
  float* o_d    = o_a    + (size_t)NA_ * H;
  float* deg_a  = o_d    + (size_t)ND_ * H;
  float* deg_d  = deg_a  + (size_t)NA_;

  // 1. x_author = author_emb[author_node_id]
  gather_author<<<(NA_ * H + 255) / 256, 256, 0, stream>>>(author_emb, author_nid, x_auth, NA_);

  // 2. x_doc = doc_x @ doc_lin_W + b
  gemm_dual<<<ND_ / 16, 256, 0, stream>>>(doc_x, doc_lin_W, nullptr, nullptr,
                                          doc_lin_b, x_doc, 0);

  // 3. layer-1 mean aggregation
  hipMemsetAsync(agg_a, 0, (size_t)NA_ * H * sizeof(float), stream);
  hipMemsetAsync(agg_d, 0, (size_t)ND_ * H * sizeof(float), stream);
  hipMemsetAsync(deg_a, 0, (size_t)NA_ * sizeof(float), stream);
  hipMemsetAsync(deg_d, 0, (size_t)ND_ * sizeof(float), stream);
  scatter_edges<<<E_, 128, 0, stream>>>(x_doc, x_auth, src, dst, agg_a, agg_d,
                                        deg_a, deg_d, E_);
  normalize_agg<<<(NA_ * H + 255) / 256, 256, 0, stream>>>(agg_a, deg_a, NA_);
  normalize_agg<<<(ND_ * H + 255) / 256, 256, 0, stream>>>(agg_d, deg_d, ND_);

  // 4. layer-1 convs + ReLU
  gemm_dual<<<NA_ / 16, 256, 0, stream>>>(agg_a, c1_da_Wl, x_auth, c1_da_Wr,
                                          c1_da_bl, h_a, 1);
  gemm_dual<<<ND_ / 16, 256, 0, stream>>>(agg_d, c1_ad_Wl, x_doc, c1_ad_Wr,
                                          c1_ad_bl, h_d, 1);

  // 5. layer-2 mean aggregation (degrees unchanged — same graph)
  hipMemsetAsync(agg_a, 0, (size_t)NA_ * H * sizeof(float), stream);
  hipMemsetAsync(agg_d, 0, (size_t)ND_ * H * sizeof(float), stream);
  scatter_edges<<<E_, 128, 0, stream>>>(h_d, h_a, src, dst, agg_a, agg_d,
                                        nullptr, nullptr, E_);
  normalize_agg<<<(NA_ * H + 255) / 256, 256, 0, stream>>>(agg_a, deg_a, NA_);
  normalize_agg<<<(ND_ * H + 255) / 256, 256, 0, stream>>>(agg_d, deg_d, ND_);

  // 6. layer-2 convs (no activation)
  gemm_dual<<<NA_ / 16, 256, 0, stream>>>(agg_a, c2_da_Wl, h_a, c2_da_Wr,
                                          c2_da_bl, o_a, 0);
  gemm_dual<<<ND_ / 16, 256, 0, stream>>>(agg_d, c2_ad_Wl, h_d, c2_ad_Wr,
                                          c2_ad_bl, o_d, 0);

  // 7. edge scorer
  score_edges<<<(EL_ + 7) / 8, 256, 0, stream>>>(o_d, o_a, eli, (float*)d_out, EL_);
}